// SoftHausdorffLoss_50981261804139
// MI455X (gfx1250) — compile-verified
//
#include <hip/hip_runtime.h>
#include <hip/hip_bf16.h>
#include <stdint.h>

// ---- CDNA5 WMMA vector types ----
typedef __attribute__((ext_vector_type(2))) float v2f;
typedef __attribute__((ext_vector_type(8))) float v8f;

#define HGRID 80
#define WGRID 80
#define NPIX  (HGRID * WGRID)          // 6400 pixels
#define MAXCOLS 6416                   // compacted-list capacity (16-padded)
#define TEMP_INV 100.0f                // 1 / TEMPERATURE
#define PAD_COORD 1000.0f              // sentinel coord -> D ~ 1.7e6 -> exp==0
#define WAVES_PER_BLOCK 4
#define TILES_PER_BATCH (NPIX / 16)    // 400 row tiles of 16
#define BLOCKS_PER_BATCH (TILES_PER_BATCH / WAVES_PER_BLOCK) // 100

// -------------------------------------------------------------------------
// Phase 1: deterministic stream-compaction of the target mask (one wave per
// batch, ballot + popc prefix). Stores (x,y) as f32 pairs; pads the list to
// a multiple of 16 with far-away sentinels whose exp underflows to 0.
// -------------------------------------------------------------------------
__global__ __launch_bounds__(32)
void compact_mask(const float* __restrict__ target,
                  int* __restrict__ cnt,
                  float2* __restrict__ coords,
                  int B) {
    int b = blockIdx.x;
    if (b >= B) return;
    int lane = threadIdx.x & 31;
    const float* t = target + (size_t)b * NPIX;
    float2* c = coords + (size_t)b * MAXCOLS;

    int base = 0;
    for (int i0 = 0; i0 < NPIX; i0 += 32) {
        int i = i0 + lane;
        bool m = t[i] > 0.5f;
        unsigned b32 = (unsigned)__ballot(m);          // wave32: low 32 bits
        int pre = __popc(b32 & ((1u << lane) - 1u));
        if (m) {
            c[base + pre] = make_float2((float)(i % WGRID),   // x
                                        (float)(i / WGRID));  // y
        }
        base += __popc(b32);
    }
    int padded = (base + 15) & ~15;
    for (int p = base + lane; p < padded; p += 32)
        c[p] = make_float2(PAD_COORD, PAD_COORD);
    if (lane == 0) cnt[b] = base;
}

// -------------------------------------------------------------------------
// Phase 2: main kernel. One wave owns a 16-row tile; inner loop walks the
// compacted target list 16 columns at a time. Cross term a.b via the
// exact-fit rank-2 instruction v_wmma_f32_16x16x4_f32:
//   A (16x4): A[m][0]=x_i, A[m][1]=y_i, K=2,3 zero (lanes 16-31 zero)
//   B (4x16): B[0][n]=x_j, B[1][n]=y_j, K=2,3 zero (lanes 16-31 zero)
// then D = |a|^2 + |b|^2 - 2ab in the WMMA C/D layout. Two passes:
// (1) per-row min, (2) exp((dmin-D)/T) sums -> exact masked softmax.
// -------------------------------------------------------------------------
__global__ __launch_bounds__(WAVES_PER_BLOCK * 32)
void soft_hausdorff_main(const float* __restrict__ pred,
                         const int* __restrict__ cnt,
                         const float2* __restrict__ coords,
                         float* __restrict__ partials) {
    __shared__ float2 sCols[MAXCOLS];   // 51.3 KB of 320 KB/WGP

    int b    = blockIdx.x / BLOCKS_PER_BATCH;
    int tgrp = blockIdx.x % BLOCKS_PER_BATCH;
    int wave = threadIdx.x >> 5;
    int lane = threadIdx.x & 31;
    int gidx = lane & 15;
    bool hi  = lane >= 16;

    int tileIdx = b * TILES_PER_BATCH + tgrp * WAVES_PER_BLOCK + wave;

    int n = cnt[b];
    if (n == 0) {                         // empty mask -> batch contributes 0
        if (lane == 0) partials[tileIdx] = 0.0f;
        return;
    }
    int nPad = (n + 15) & ~15;

    // cooperative stage of the compacted coord list into LDS
    const float2* g = coords + (size_t)b * MAXCOLS;
    for (int i = threadIdx.x; i < nPad; i += blockDim.x) sCols[i] = g[i];
    __syncthreads();

    int mBase = (tgrp * WAVES_PER_BLOCK + wave) * 16;

    // A operand (32-bit A 16x4): lanes 0-15 hold M=lane, VGPR0=K0(x), VGPR1=K1(y);
    // lanes 16-31 hold K=2,3 -> zero.
    v2f aOp = {};
    if (!hi) {
        int row = mBase + gidx;
        aOp[0] = (float)(row % WGRID);
        aOp[1] = (float)(row / WGRID);
    }

    // |a|^2 for the 8 rows this lane owns in the C/D layout:
    // VGPR r, lanes 0-15 -> row mBase+r ; lanes 16-31 -> row mBase+8+r
    float sqi[8];
#pragma unroll
    for (int r = 0; r < 8; ++r) {
        int row = mBase + r + (hi ? 8 : 0);
        float x = (float)(row % WGRID), y = (float)(row / WGRID);
        sqi[r] = x * x + y * y;
    }

    // ---------------- pass 1: row-wise min distance ----------------
    float dmin[8];
#pragma unroll
    for (int r = 0; r < 8; ++r) dmin[r] = 3.0e38f;

    for (int nb = 0; nb < nPad; nb += 16) {
        float2 p = sCols[nb + gidx];
        float sqj = p.x * p.x + p.y * p.y;
        v2f bOp = {};
        if (!hi) { bOp[0] = p.x; bOp[1] = p.y; }
        v8f c = {};
        v8f ab = __builtin_amdgcn_wmma_f32_16x16x4_f32(
            false, aOp, false, bOp, (short)0, c, false, false);
#pragma unroll
        for (int r = 0; r < 8; ++r) {
            float d = fmaxf(sqi[r] + sqj - 2.0f * ab[r], 0.0f);
            dmin[r] = fminf(dmin[r], d);
        }
    }
    // min-reduce the 16 lane-partials of each row (stays within 16-lane half)
#pragma unroll
    for (int r = 0; r < 8; ++r) {
        float v = dmin[r];
        v = fminf(v, __shfl_xor(v, 1));
        v = fminf(v, __shfl_xor(v, 2));
        v = fminf(v, __shfl_xor(v, 4));
        v = fminf(v, __shfl_xor(v, 8));
        dmin[r] = v;
    }

    // ---------------- pass 2: exp sums with known row max ----------------
    float S[8], W[8];
#pragma unroll
    for (int r = 0; r < 8; ++r) { S[r] = 0.0f; W[r] = 0.0f; }

    for (int nb = 0; nb < nPad; nb += 16) {
        float2 p = sCols[nb + gidx];
        float sqj = p.x * p.x + p.y * p.y;
        v2f bOp = {};
        if (!hi) { bOp[0] = p.x; bOp[1] = p.y; }
        v8f c = {};
        v8f ab = __builtin_amdgcn_wmma_f32_16x16x4_f32(
            false, aOp, false, bOp, (short)0, c, false, false);
#pragma unroll
        for (int r = 0; r < 8; ++r) {
            float d = fmaxf(sqi[r] + sqj - 2.0f * ab[r], 0.0f);
            float e = __expf((dmin[r] - d) * TEMP_INV);  // pads -> exp(-2e8)=0
            S[r] += e;
            W[r] += e * d;
        }
    }
    // sum-reduce across the 16 lane-partials of each row
#pragma unroll
    for (int r = 0; r < 8; ++r) {
        float s = S[r], w = W[r];
        s += __shfl_xor(s, 1); w += __shfl_xor(w, 1);
        s += __shfl_xor(s, 2); w += __shfl_xor(w, 2);
        s += __shfl_xor(s, 4); w += __shfl_xor(w, 4);
        s += __shfl_xor(s, 8); w += __shfl_xor(w, 8);
        S[r] = s; W[r] = w;
    }

    // lanes 0 and 16 each own 8 distinct rows; combine and emit one partial
    float part = 0.0f;
    if (gidx == 0) {
        const float* p = pred + (size_t)b * NPIX;
#pragma unroll
        for (int r = 0; r < 8; ++r) {
            int row = mBase + r + (hi ? 8 : 0);
            part += p[row] * (W[r] / S[r]);   // S >= 1 (argmin column -> e=1)
        }
    }
    part += __shfl_xor(part, 16);
    if (lane == 0) partials[tileIdx] = part;
}

// -------------------------------------------------------------------------
// Phase 3: deterministic fixed-order reduction of per-tile partials.
// -------------------------------------------------------------------------
__global__ __launch_bounds__(256)
void reduce_partials(const float* __restrict__ partials, int n,
                     float* __restrict__ out, float invB) {
    __shared__ float s[256];
    float acc = 0.0f;
    for (int i = threadIdx.x; i < n; i += 256) acc += partials[i];
    s[threadIdx.x] = acc;
    __syncthreads();
    for (int off = 128; off > 0; off >>= 1) {
        if (threadIdx.x < (unsigned)off) s[threadIdx.x] += s[threadIdx.x + off];
        __syncthreads();
    }
    if (threadIdx.x == 0) out[0] = s[0] * invB;
}

extern "C" void kernel_launch(void* const* d_in, const int* in_sizes, int n_in,
                              void* d_out, int out_size, void* d_ws, size_t ws_size,
                              hipStream_t stream) {
    const float* pred   = (const float*)d_in[0];
    const float* target = (const float*)d_in[1];
    float* out = (float*)d_out;
    int B = in_sizes[0] / NPIX;           // pred is (B,1,80,80) f32

    // workspace layout
    char* ws = (char*)d_ws;
    int* cnt = (int*)ws;
    size_t off = 256;
    float2* coords = (float2*)(ws + off);
    off += (size_t)B * MAXCOLS * sizeof(float2);
    off = (off + 255) & ~(size_t)255;
    float* partials = (float*)(ws + off);

    compact_mask<<<B, 32, 0, stream>>>(target, cnt, coords, B);
    soft_hausdorff_main<<<B * BLOCKS_PER_BATCH, WAVES_PER_BLOCK * 32, 0, stream>>>(
        pred, cnt, coords, partials);
    reduce_partials<<<1, 256, 0, stream>>>(partials, B * TILES_PER_BATCH, out,
                                           1.0f / (float)B);
}